// RNNModel_42563125903898
// MI455X (gfx1250) — compile-verified
//
#include <hip/hip_runtime.h>
#include <math.h>

// ---------------- problem constants ----------------
#define NTOK   50000
#define NINP   256
#define NHID   256
#define SEQL   128
#define BSZ    64
#define NS     32
#define SB     (SEQL * BSZ)          // 8192
#define TEMP_  65.0f

typedef float v2f __attribute__((ext_vector_type(2)));
typedef float v8f __attribute__((ext_vector_type(8)));

// ---------------- f32 WMMA 16x16x4 (codegen-confirmed round 1) -------------
// D = A(16x4)*B(4x16) + C(16x16), wave32.
// A: lane -> M=lane%16 ; lanes 0-15 hold K={0,1}, lanes 16-31 hold K={2,3}.
// B (mirrored): lane -> N=lane%16 ; same K split.
// C/D: VGPR r -> M = r + (lane<16?0:8), N = lane%16.
__device__ __forceinline__ v8f wmma_f32_4(v2f a, v2f b, v8f c) {
  return __builtin_amdgcn_wmma_f32_16x16x4_f32(false, a, false, b,
                                               (short)0, c, false, false);
}

#define V8Z {0.f,0.f,0.f,0.f,0.f,0.f,0.f,0.f}

// A-operand swizzle position (element [m][k] of a 64x256 matrix):
//   pos = ((m>>4)*64 + (k>>2))*32 + (m&15) + ((k&2)?16:0), sub = k&1
__device__ __forceinline__ int aswz(int m, int k) {
  return ((((m >> 4) * 64 + (k >> 2)) * 32) + (m & 15) + ((k & 2) ? 16 : 0)) * 2 + (k & 1);
}

// =====================================================================
// Kernel 1: C[m,n] = sum_k A[m,k]*W[n,k] + bias[n]   (A: Mx256, W: 256x256)
// One wave -> 32(M) x 16(N) tile: B loads shared by 2 M-tiles,
// 4 independent accumulators for WMMA ILP.  Used for embW (M=50000).
// =====================================================================
__global__ void __launch_bounds__(256)
gemm_aWt_bias(const float* __restrict__ A, const float* __restrict__ W,
              const float* __restrict__ bias, float* __restrict__ C, int M) {
  const int wave = blockIdx.x * 8 + (threadIdx.x >> 5);
  const int lane = threadIdx.x & 31;
  const int mT2  = wave >> 4;                 // 32-row tile index
  const int nT   = wave & 15;
  const int mloc = lane & 15;
  const int koff = (lane < 16) ? 0 : 2;
  const int n    = nT * 16 + mloc;

  const int m0 = mT2 * 32 + mloc;
  const int m1 = m0 + 16;
  const int m0c = (m0 < M) ? m0 : (M - 1);    // clamp loads, keep EXEC all-1s
  const int m1c = (m1 < M) ? m1 : (M - 1);

  v8f c00 = V8Z, c01 = V8Z, c10 = V8Z, c11 = V8Z;
  const float* a0 = A + (size_t)m0c * 256 + koff;
  const float* a1 = A + (size_t)m1c * 256 + koff;
  const float* wr = W + (size_t)n * 256 + koff;

#pragma unroll 4
  for (int k = 0; k < 256; k += 8) {
    v2f b0  = *(const v2f*)(wr + k);
    v2f b1  = *(const v2f*)(wr + k + 4);
    v2f x00 = *(const v2f*)(a0 + k);
    v2f x01 = *(const v2f*)(a0 + k + 4);
    v2f x10 = *(const v2f*)(a1 + k);
    v2f x11 = *(const v2f*)(a1 + k + 4);
    c00 = wmma_f32_4(x00, b0, c00);
    c10 = wmma_f32_4(x10, b0, c10);
    c01 = wmma_f32_4(x01, b1, c01);
    c11 = wmma_f32_4(x11, b1, c11);
  }

  const float bv = bias[n];
  const int mb0 = mT2 * 32 + ((lane < 16) ? 0 : 8);
  for (int r = 0; r < 8; ++r) {
    const int row0 = mb0 + r, row1 = row0 + 16;
    if (row0 < M) C[(size_t)row0 * 256 + n] = c00[r] + c01[r] + bv;
    if (row1 < M) C[(size_t)row1 * 256 + n] = c10[r] + c11[r] + bv;
  }
}

// =====================================================================
// Kernel 2: xW[n,:] = embW[data[n],:] + b_hh   (one block per row)
// (embW already holds emb@W_ih^T + b_ih, so xW = full input-side preactivation)
// =====================================================================
__global__ void __launch_bounds__(256)
gather_xw(const int* __restrict__ data, const float* __restrict__ embW,
          const float* __restrict__ b_hh, float* __restrict__ xW) {
  const int n = blockIdx.x;
  const int k = threadIdx.x;
  const int tok = data[n];
  xW[(size_t)n * NHID + k] = embW[(size_t)tok * NHID + k] + b_hh[k];
}

// =====================================================================
// Kernel 3: pre-swizzle W_hh into B-operand layout:
//   WhhB[((nt*64+ks)*32+lane)*2 + {0,1}] = W_hh[nt*16 + lane%16][ks*4 + (lane<16?0:2) + {0,1}]
// so the RNN inner loop is one coalesced global_load_b64 per k-step.
// =====================================================================
__global__ void __launch_bounds__(256)
prep_whhB(const float* __restrict__ W_hh, float* __restrict__ WhhB) {
  const int i    = blockIdx.x * 256 + threadIdx.x;   // 0..32767
  const int lane = i & 31;
  const int ks   = (i >> 5) & 63;
  const int nt   = i >> 11;
  const int nrow = nt * 16 + (lane & 15);
  const int k    = ks * 4 + ((lane < 16) ? 0 : 2);
  v2f v;
  v[0] = W_hh[(size_t)nrow * 256 + k];
  v[1] = W_hh[(size_t)nrow * 256 + k + 1];
  *(v2f*)(WhhB + (size_t)i * 2) = v;
}

// =====================================================================
// Kernel 4: sequential RNN.  1 block x 1024 threads (32 waves).
//   h_t = tanh(xW_t + h_{t-1} @ W_hh^T)
// H state kept in LDS in the A-operand swizzled layout (conflict-free
// ds_load_b64 per k-step).  Also emits hU = acc + b_hh for free.
// =====================================================================
__global__ void __launch_bounds__(1024)
rnn_forward(const float* __restrict__ hidden, const float* __restrict__ xW,
            const float* __restrict__ WhhB, const float* __restrict__ b_hh,
            float* __restrict__ ro, float* __restrict__ hU) {
  __shared__ float HA[BSZ * NHID];              // 64 KB, A-swizzled
  const int tid  = threadIdx.x;
  const int wave = tid >> 5;
  const int lane = tid & 31;
  const int mloc = lane & 15;
  const int rbas = (lane < 16) ? 0 : 8;

  // init: ro[0] = hidden ; HA = swizzle(hidden)
  for (int i = tid; i < BSZ * NHID; i += 1024) {
    const float v = hidden[i];
    ro[i] = v;
    HA[aswz(i >> 8, i & 255)] = v;
  }
  __syncthreads();

  const int tile0 = wave * 2;
  float bh[2];
  for (int tt = 0; tt < 2; ++tt)
    bh[tt] = b_hh[((tile0 + tt) & 15) * 16 + mloc];

  for (int t = 0; t < SEQL; ++t) {
    int kb = 0;
    asm volatile("" : "+s"(kb));                // opaque: blocks LICM/spill of W loads
    const float* wB = WhhB + kb;
    float hout[2][8];

    for (int tt = 0; tt < 2; ++tt) {
      const int tile = tile0 + tt;
      const int mT = tile >> 4;                 // 0..3
      const int nT = tile & 15;                 // 0..15
      const int n  = nT * 16 + mloc;

      v8f c0 = V8Z, c1 = V8Z;
      const float* arow = HA + (size_t)mT * 4096 + lane * 2;  // LDS, conflict-free
      const float* brow = wB + (size_t)nT * 4096 + lane * 2;  // global, coalesced
#pragma unroll 4
      for (int ks = 0; ks < 64; ks += 2) {
        v2f a0 = *(const v2f*)(arow + ks * 64);
        v2f b0 = *(const v2f*)(brow + ks * 64);
        c0 = wmma_f32_4(a0, b0, c0);
        v2f a1 = *(const v2f*)(arow + (ks + 1) * 64);
        v2f b1 = *(const v2f*)(brow + (ks + 1) * 64);
        c1 = wmma_f32_4(a1, b1, c1);
      }

      const int mb = mT * 16 + rbas;
      const float* xrow  = xW + ((size_t)t * BSZ + mb) * NHID + n;
      float*       hUrow = hU + ((size_t)t * BSZ + mb) * NHID + n;
      for (int r = 0; r < 8; ++r) {
        const float g = c0[r] + c1[r];          // = (h_{t-1} @ W_hh^T)[mb+r][n]
        hUrow[(size_t)r * NHID] = g + bh[tt];   // hU row (t*64+mb+r)
        hout[tt][r] = tanhf(g + xrow[(size_t)r * NHID]);
      }
    }

    __syncthreads();                            // all reads of old H done
    float* rot = ro + (size_t)(t + 1) * BSZ * NHID;
    for (int tt = 0; tt < 2; ++tt) {
      const int tile = tile0 + tt;
      const int mT = tile >> 4;
      const int nT = tile & 15;
      const int n  = nT * 16 + mloc;
      const int mb = mT * 16 + rbas;
      for (int r = 0; r < 8; ++r) {
        const float h = hout[tt][r];
        const int m = mb + r;
        rot[(size_t)m * NHID + n] = h;
        HA[aswz(m, n)] = h;
      }
    }
    __syncthreads();                            // new H visible
  }
}

// =====================================================================
// Kernel 5: fused distances.  One wave per n.
//   x[0,n]   = 65*(-||ro[n]-ro[n+64]||^2 + bias[data[n]])
//   x[1+s,n] = 65*(-||h_prev[n]-tanh(embW[tok]+hU[n])||^2 + bias[tok])
// embW (51.2 MB) is L2-resident on MI455X (192 MB L2).
// =====================================================================
__global__ void __launch_bounds__(256)
neg_dist(const int* __restrict__ data, const int* __restrict__ samples,
         const float* __restrict__ bias_vec, const float* __restrict__ ro,
         const float* __restrict__ embW, const float* __restrict__ hU,
         float* __restrict__ x) {
  const int n = blockIdx.x * (blockDim.x >> 5) + (threadIdx.x >> 5);
  const int lane = threadIdx.x & 31;
  if (n >= SB) return;

  const float* hp = ro + (size_t)n * NHID;
  float hpv[8], huv[8];
  for (int i = 0; i < 8; ++i) hpv[i] = hp[lane + i * 32];

  const float* hn = ro + (size_t)(n + BSZ) * NHID;   // h_t row
  float d = 0.0f;
  for (int i = 0; i < 8; ++i) {
    const float df = hpv[i] - hn[lane + i * 32];
    d += df * df;
  }
  for (int off = 16; off; off >>= 1) d += __shfl_xor(d, off, 32);
  if (lane == 0) x[n] = TEMP_ * (-d + bias_vec[data[n]]);

  const float* hu = hU + (size_t)n * NHID;
  for (int i = 0; i < 8; ++i) huv[i] = hu[lane + i * 32];

  for (int s = 0; s < NS; ++s) {
    const int tok = samples[s * SB + n];
    const float* ew = embW + (size_t)tok * NHID;
    float dd = 0.0f;
    for (int i = 0; i < 8; ++i) {
      const float v  = tanhf(ew[lane + i * 32] + huv[i]);
      const float df = hpv[i] - v;
      dd += df * df;
    }
    for (int off = 16; off; off >>= 1) dd += __shfl_xor(dd, off, 32);
    if (lane == 0) x[(size_t)(s + 1) * SB + n] = TEMP_ * (-dd + bias_vec[tok]);
  }
}

// =====================================================================
// Kernel 6: per-column  logsumexp - positive logit,  block partial sums.
// =====================================================================
__global__ void __launch_bounds__(256)
lse_partial(const float* __restrict__ x, float* __restrict__ partial) {
  const int n = blockIdx.x * blockDim.x + threadIdx.x;
  float v = 0.0f;
  if (n < SB) {
    float mx = x[n];
    for (int s = 1; s < NS + 1; ++s) mx = fmaxf(mx, x[(size_t)s * SB + n]);
    float sum = 0.0f;
    for (int s = 0; s < NS + 1; ++s) sum += expf(x[(size_t)s * SB + n] - mx);
    v = (mx + logf(sum)) - x[n];                 // -log_softmax[0]
  }
  __shared__ float red[256];
  red[threadIdx.x] = v;
  __syncthreads();
  for (int st = 128; st; st >>= 1) {
    if (threadIdx.x < st) red[threadIdx.x] += red[threadIdx.x + st];
    __syncthreads();
  }
  if (threadIdx.x == 0) partial[blockIdx.x] = red[0];
}

// =====================================================================
// Kernel 7: finalize  loss = mean + sum(bias^2)   (deterministic trees)
// =====================================================================
__global__ void __launch_bounds__(256)
finalize_loss(const float* __restrict__ partial, const float* __restrict__ bias_vec,
              float* __restrict__ loss_out, int nPartials) {
  __shared__ float red[256];
  const int tid = threadIdx.x;
  float b2 = 0.0f;
  for (int i = tid; i < NTOK; i += 256) { const float b = bias_vec[i]; b2 += b * b; }
  red[tid] = b2;
  __syncthreads();
  for (int st = 128; st; st >>= 1) {
    if (tid < st) red[tid] += red[tid + st];
    __syncthreads();
  }
  const float biassum = red[0];
  __syncthreads();
  red[tid] = (tid < nPartials) ? partial[tid] : 0.0f;
  __syncthreads();
  for (int st = 128; st; st >>= 1) {
    if (tid < st) red[tid] += red[tid + st];
    __syncthreads();
  }
  if (tid == 0) loss_out[0] = red[0] / (float)SB + biassum;
}

// =====================================================================
// launch
// =====================================================================
extern "C" void kernel_launch(void* const* d_in, const int* in_sizes, int n_in,
                              void* d_out, int out_size, void* d_ws, size_t ws_size,
                              hipStream_t stream) {
  const int*   data      = (const int*)  d_in[0];
  const float* hidden    = (const float*)d_in[1];
  const int*   samples   = (const int*)  d_in[2];
  const float* emb_table = (const float*)d_in[3];
  const float* W_ih      = (const float*)d_in[4];
  const float* b_ih      = (const float*)d_in[5];
  const float* W_hh      = (const float*)d_in[6];
  const float* b_hh      = (const float*)d_in[7];
  const float* bias_vec  = (const float*)d_in[8];
  float* out = (float*)d_out;             // [0]=loss, [1..16384]=new_hidden

  // workspace layout (floats): ~77.8 MB total
  float* ro   = (float*)d_ws;                             // 8256 x 256
  float* embW = ro   + (size_t)(SEQL + 1) * BSZ * NHID;   // 50000 x 256
  float* xW   = embW + (size_t)NTOK * NHID;               // 8192 x 256
  float* hU   = xW   + (size_t)SB * NHID;                 // 8192 x 256
  float* WhhB = hU   + (size_t)SB * NHID;                 // 65536
  float* x    = WhhB + (size_t)16 * 64 * 32 * 2;          // 33 x 8192
  float* part = x    + (size_t)(NS + 1) * SB;             // 32

  // 1) embW[t] = emb_table[t] @ W_ih^T + b_ih  (6.55 GFLOP, L2-resident result)
  const int mT2 = (NTOK + 31) / 32;
  gemm_aWt_bias<<<mT2 * 2, 256, 0, stream>>>(emb_table, W_ih, b_ih, embW, NTOK);

  // 2) xW[n] = embW[data[n]] + b_hh  (input-side preactivation for all steps)
  gather_xw<<<SB, 256, 0, stream>>>(data, embW, b_hh, xW);

  // 3) W_hh -> B-operand swizzled layout
  prep_whhB<<<128, 256, 0, stream>>>(W_hh, WhhB);

  // 4) sequential RNN (also emits hU = h_prev@W_hh^T + b_hh for free)
  rnn_forward<<<1, 1024, 0, stream>>>(hidden, xW, WhhB, b_hh, ro, hU);

  // 5) fused tanh + squared-distance logits (33 x 8192)
  neg_dist<<<SB / 8, 256, 0, stream>>>(data, samples, bias_vec, ro, embW, hU, x);

  // 6) deterministic loss reduction
  lse_partial<<<SB / 256, 256, 0, stream>>>(x, part);
  finalize_loss<<<1, 256, 0, stream>>>(part, bias_vec, out, SB / 256);

  // 7) new_hidden = ro[-1] (rows 8192..8255)
  hipMemcpyAsync(out + 1, ro + (size_t)SB * NHID, (size_t)BSZ * NHID * sizeof(float),
                 hipMemcpyDeviceToDevice, stream);
}